// LandmarkModel_49469433315727
// MI455X (gfx1250) — compile-verified
//
#include <hip/hip_runtime.h>
#include <stdint.h>

// ---------------------------------------------------------------------------
// LandmarkModel: dedup-scatter (bitmask) + popcount reduce + streaming gather.
// out[i] = (counts[idx[i]] + 1) / (obs_count + #unique(idx))
// presence[idx[i]] == 1 by construction for every gathered position, so the
// scatter result is only needed for the scalar denominator.
// ---------------------------------------------------------------------------

#define MASK_WORDS_FOR(nl) (((nl) + 31) / 32)
#define MASK_OFFSET_BYTES 256   // ws layout: [0..255] control, then bitmask

// ---- CDNA5 async global->LDS support (guarded; fallback = direct loads) ----
#if defined(__has_builtin)
#  if __has_builtin(__builtin_amdgcn_global_load_async_to_lds_b128)
#    define HAVE_ASYNC_LDS 1
#  else
#    define HAVE_ASYNC_LDS 0
#  endif
#  if __has_builtin(__builtin_amdgcn_s_wait_asynccnt)
#    define HAVE_WAIT_BUILTIN 1
#  else
#    define HAVE_WAIT_BUILTIN 0
#  endif
#else
#  define HAVE_ASYNC_LDS 0
#  define HAVE_WAIT_BUILTIN 0
#endif

#if HAVE_WAIT_BUILTIN
#  define WAIT_ASYNCCNT(n) do { __builtin_amdgcn_s_wait_asynccnt(n); \
                                asm volatile("" ::: "memory"); } while (0)
#else
#  define WAIT_ASYNCCNT(n) asm volatile("s_wait_asynccnt %0" :: "i"(n) : "memory")
#endif

// clang native vector types (required by nontemporal builtins; HIP_vector_type
// structs are rejected)
typedef int   v4i __attribute__((ext_vector_type(4)));
typedef float v4f __attribute__((ext_vector_type(4)));
typedef __attribute__((address_space(1))) v4i gv4i;   // global
typedef __attribute__((address_space(3))) v4i lv4i;   // LDS

#if HAVE_ASYNC_LDS
__device__ __forceinline__ void async_copy_b128(const int* gsrc, int* ldst) {
    // per-lane: copies 16 bytes from global address gsrc to LDS address ldst, tracked by ASYNCcnt
    __builtin_amdgcn_global_load_async_to_lds_b128(
        (gv4i*)(uintptr_t)gsrc, (lv4i*)ldst, 0, 0);
}
#endif

// ---------------------------------------------------------------------------
// Kernel 1: zero the presence bitmask (+ control slot)
// ---------------------------------------------------------------------------
__global__ void k_init(unsigned int* __restrict__ mask, float* __restrict__ ctrl,
                       int maskWords) {
    int i = blockIdx.x * blockDim.x + threadIdx.x;
    if (i < maskWords) mask[i] = 0u;
    if (i == 0) ctrl[0] = 0.0f;
}

// ---------------------------------------------------------------------------
// Kernel 2: OR-scatter presence bits. Pre-check through (possibly stale)
// cache before the atomic: safe because OR is idempotent; cuts L2 atomics
// from 16.7M to ~#unique. Index stream is read-once -> non-temporal.
// ---------------------------------------------------------------------------
__global__ void k_scatter(const int* __restrict__ idx,
                          unsigned int* __restrict__ mask, int n) {
    const int stride = gridDim.x * blockDim.x;
    const int n4 = n >> 2;
    const v4i* __restrict__ idx4 = (const v4i*)idx;
    for (int i = blockIdx.x * blockDim.x + threadIdx.x; i < n4; i += stride) {
        __builtin_prefetch(idx4 + i + stride, 0, 0);  // global_prefetch_b8
        v4i v = __builtin_nontemporal_load(idx4 + i);
        int as[4] = {v.x, v.y, v.z, v.w};
#pragma unroll
        for (int j = 0; j < 4; ++j) {
            unsigned int w = ((unsigned int)as[j]) >> 5;
            unsigned int b = 1u << (as[j] & 31);
            if ((mask[w] & b) == 0u) atomicOr(&mask[w], b);
        }
    }
    // tail (n not multiple of 4)
    for (int i = (n4 << 2) + blockIdx.x * blockDim.x + threadIdx.x; i < n;
         i += stride) {
        unsigned int w = ((unsigned int)idx[i]) >> 5;
        unsigned int b = 1u << (idx[i] & 31);
        if ((mask[w] & b) == 0u) atomicOr(&mask[w], b);
    }
}

// ---------------------------------------------------------------------------
// Kernel 3: popcount-reduce the bitmask (single workgroup), compute
// inv_total = 1 / (obs_count + #unique)
// ---------------------------------------------------------------------------
__global__ void k_reduce(const unsigned int* __restrict__ mask, int maskWords,
                         const float* __restrict__ obs, float* __restrict__ ctrl) {
    __shared__ unsigned int sred[1024];
    unsigned int s = 0;
    for (int i = threadIdx.x; i < maskWords; i += blockDim.x)
        s += (unsigned int)__popc(mask[i]);
    sred[threadIdx.x] = s;
    __syncthreads();
    for (int off = blockDim.x >> 1; off > 0; off >>= 1) {
        if ((int)threadIdx.x < off) sred[threadIdx.x] += sred[threadIdx.x + off];
        __syncthreads();
    }
    if (threadIdx.x == 0) {
        float total = obs[0] + (float)sred[0];
        ctrl[0] = 1.0f / total;
    }
}

// ---------------------------------------------------------------------------
// Kernel 4: streaming gather. Per-wave double-buffered async global->LDS
// pipeline for the index stream (chunk = 128 indices = 512B = wave32 x b128),
// then ds_load_b128 + 4 gathered counts (L2-resident) + non-temporal b128
// store of the output stream.
// Buffer-reuse hazard is safe: the ds_load result is consumed (forcing
// s_wait_dscnt 0) before the next async issue targets that buffer.
// ---------------------------------------------------------------------------
__global__ __launch_bounds__(256) void k_gather(const int* __restrict__ idx,
                                                const float* __restrict__ counts,
                                                const float* __restrict__ ctrl,
                                                float* __restrict__ out,
                                                int nChunks, int n) {
    __shared__ int smem[8 * 2 * 128];  // 8 waves x 2 buffers x 128 ints = 8KB
    const int tid = threadIdx.x;
    const int lane = tid & 31;
    const int wave = tid >> 5;
    const int wavesTotal = gridDim.x * 8;
    const int waveGlobal = blockIdx.x * 8 + wave;
    int* __restrict__ wbuf = smem + wave * 256;  // two 128-int buffers
    const float inv = ctrl[0];

#if HAVE_ASYNC_LDS
    int c = waveGlobal;
    if (c < nChunks) {
        async_copy_b128(idx + (size_t)c * 128 + lane * 4, wbuf + lane * 4);
        int buf = 0;
        for (; c < nChunks; c += wavesTotal) {
            const int cn = c + wavesTotal;
            if (cn < nChunks) {
                async_copy_b128(idx + (size_t)cn * 128 + lane * 4,
                                wbuf + ((buf ^ 1) * 128) + lane * 4);
                WAIT_ASYNCCNT(1);   // oldest (current chunk) complete
            } else {
                WAIT_ASYNCCNT(0);
            }
            v4i v = *(const v4i*)(wbuf + buf * 128 + lane * 4);
            v4f r;
            r.x = (counts[v.x] + 1.0f) * inv;
            r.y = (counts[v.y] + 1.0f) * inv;
            r.z = (counts[v.z] + 1.0f) * inv;
            r.w = (counts[v.w] + 1.0f) * inv;
            __builtin_nontemporal_store(r, (v4f*)(out + (size_t)c * 128 + lane * 4));
            buf ^= 1;
        }
    }
#else
    // Fallback: direct vectorized loads (no async path available)
    for (int c = waveGlobal; c < nChunks; c += wavesTotal) {
        v4i v = *(const v4i*)(idx + (size_t)c * 128 + lane * 4);
        v4f r;
        r.x = (counts[v.x] + 1.0f) * inv;
        r.y = (counts[v.y] + 1.0f) * inv;
        r.z = (counts[v.z] + 1.0f) * inv;
        r.w = (counts[v.w] + 1.0f) * inv;
        __builtin_nontemporal_store(r, (v4f*)(out + (size_t)c * 128 + lane * 4));
    }
#endif
    // tail (n not multiple of 128)
    const int tailStart = nChunks * 128;
    const int gstride = gridDim.x * blockDim.x;
    for (int i = tailStart + blockIdx.x * blockDim.x + tid; i < n; i += gstride) {
        float r = (counts[idx[i]] + 1.0f) * inv;
        __builtin_nontemporal_store(r, out + i);
    }
}

// ---------------------------------------------------------------------------
// Launcher. Inputs: [0]=counts f32[1M], [1]=obs_count f32[1], [2]=indices i32[16.7M]
// Output: f32[16.7M]. ws: [0..255] control floats, then bitmask (125KB).
// ---------------------------------------------------------------------------
extern "C" void kernel_launch(void* const* d_in, const int* in_sizes, int n_in,
                              void* d_out, int out_size, void* d_ws, size_t ws_size,
                              hipStream_t stream) {
    const float* counts = (const float*)d_in[0];
    const float* obs    = (const float*)d_in[1];
    const int*   lidx   = (const int*)d_in[2];
    float*       out    = (float*)d_out;

    const int numLandmarks = in_sizes[0];
    const int n            = in_sizes[2];
    const int maskWords    = MASK_WORDS_FOR(numLandmarks);

    float* ctrl        = (float*)d_ws;
    unsigned int* mask = (unsigned int*)((char*)d_ws + MASK_OFFSET_BYTES);

    (void)n_in; (void)out_size; (void)ws_size;

    // 1) zero bitmask
    k_init<<<(maskWords + 255) / 256, 256, 0, stream>>>(mask, ctrl, maskWords);
    // 2) dedup OR-scatter
    k_scatter<<<2048, 256, 0, stream>>>(lidx, mask, n);
    // 3) popcount reduce -> inv_total
    k_reduce<<<1, 1024, 0, stream>>>(mask, maskWords, obs, ctrl);
    // 4) streaming gather with async global->LDS double buffering
    const int nChunks = n / 128;
    k_gather<<<1024, 256, 0, stream>>>(lidx, counts, ctrl, out, nChunks, n);
}